// SparsifyAttention_74741020885070
// MI455X (gfx1250) — compile-verified
//
#include <hip/hip_runtime.h>
#include <stdint.h>

#ifndef __has_builtin
#define __has_builtin(x) 0
#endif

#define CROW 192           // row length
#define WPB  8             // waves per block
#define BLOCK_T (WPB * 32) // 256 threads, 8 wave32s

#if defined(__HIP_DEVICE_COMPILE__) && __has_builtin(__builtin_amdgcn_ballot_w32)
#define BALLOT32(p) __builtin_amdgcn_ballot_w32(p)
#else
#define BALLOT32(p) ((unsigned)__ballot(p))
#endif

// ---- CDNA5 async global->LDS copies via inline asm (bypasses builtin type
// rules; mnemonics per ISA §15.18 / gfx1250 MC syntax). The LDS destination
// VGPR holds the low 32 bits of the generic shared pointer == in-allocation
// LDS byte offset per the CDNA5 aperture rules. ----------------------------
__device__ __forceinline__ void async_copy_b128(const float* gsrc, const float* ldst) {
  unsigned lds = (unsigned)(unsigned long long)(const void*)ldst;
  unsigned long long ga = (unsigned long long)gsrc;
  asm volatile("global_load_async_to_lds_b128 %0, %1, off"
               :: "v"(lds), "v"(ga) : "memory");
}
__device__ __forceinline__ void async_copy_b64(const float* gsrc, const float* ldst) {
  unsigned lds = (unsigned)(unsigned long long)(const void*)ldst;
  unsigned long long ga = (unsigned long long)gsrc;
  asm volatile("global_load_async_to_lds_b64 %0, %1, off"
               :: "v"(lds), "v"(ga) : "memory");
}
__device__ __forceinline__ void wait_async_le2() {
  asm volatile("s_wait_asynccnt 2" ::: "memory");
}
__device__ __forceinline__ void wait_async_le0() {
  asm volatile("s_wait_asynccnt 0" ::: "memory");
}

// Monotone float->u32 key: order(key) == order(float), exact.
__device__ __forceinline__ uint32_t fkey(float f) {
  uint32_t u = __float_as_uint(f);
  return u ^ ((uint32_t)((int32_t)u >> 31) | 0x80000000u);
}

// Process one 192-float row held as 6 values/lane across a wave32.
// a[0..3] = elements 4*lane .. 4*lane+3 ; a[4..5] = elements 128+2*lane..+1
__device__ __forceinline__ void process_row(const float a[6],
                                            float* __restrict__ outp,
                                            int lane,
                                            float w1, float w2, float w3, float w4) {
  uint32_t key[6];
#pragma unroll
  for (int r = 0; r < 6; ++r) key[r] = fkey(a[r]);

  // Row max (for exp) and min (search seed): in-lane then 5-step xor-shuffle.
  float mx = a[0], mn = a[0];
#pragma unroll
  for (int r = 1; r < 6; ++r) { mx = fmaxf(mx, a[r]); mn = fminf(mn, a[r]); }
#pragma unroll
  for (int m = 16; m > 0; m >>= 1) {
    mx = fmaxf(mx, __shfl_xor(mx, m, 32));
    mn = fminf(mn, __shfl_xor(mn, m, 32));
  }

  // --- 4 interleaved ballot-bisection selects on integer keys -------------
  const uint32_t kq[4] = {96u, 128u, 144u, 153u};
  const uint32_t l0 = fkey(mn);
  const uint32_t h0 = fkey(mx) + 1u; // count(>=h0)==0 ; finite floats => no overflow
  uint32_t lo[4], hi[4], thr[4];
  bool done[4];
#pragma unroll
  for (int s = 0; s < 4; ++s) { lo[s] = l0; hi[s] = h0; thr[s] = l0; done[s] = false; }

  int remaining = 4;
  for (int it = 0; it < 34 && remaining > 0; ++it) {
#pragma unroll
    for (int s = 0; s < 4; ++s) {
      if (done[s]) continue;
      uint32_t mid = lo[s] + ((hi[s] - lo[s]) >> 1);
      uint32_t c = 0;
#pragma unroll
      for (int r = 0; r < 6; ++r) c += (uint32_t)__popc(BALLOT32(key[r] >= mid)); // v_cmp + s_bcnt1
      c = (uint32_t)__builtin_amdgcn_readfirstlane((int)c);
      if (c >= kq[s]) {
        lo[s] = mid;
        if (c == kq[s]) { thr[s] = mid; done[s] = true; --remaining; continue; } // exact separator
      } else {
        hi[s] = mid;
      }
      if (hi[s] - lo[s] == 1u) { thr[s] = lo[s]; done[s] = true; --remaining; } // duplicate-safe exact k-th value
    }
  }

  // --- shared-max exp, masked sums, nested blend --------------------------
  float e[6];
#pragma unroll
  for (int r = 0; r < 6; ++r) e[r] = __expf(a[r] - mx);

  float Z[4];
#pragma unroll
  for (int s = 0; s < 4; ++s) {
    float p = 0.f;
#pragma unroll
    for (int r = 0; r < 6; ++r) p += (key[r] >= thr[s]) ? e[r] : 0.f;
#pragma unroll
    for (int m = 16; m > 0; m >>= 1) p += __shfl_xor(p, m, 32);
    Z[s] = p;
  }

  const float q0 = w1 / Z[0], q1 = w2 / Z[1], q2 = w3 / Z[2], q3 = w4 / Z[3];
  const float Q3 = q3, Q2 = q3 + q2, Q1 = Q2 + q1, Q0 = Q1 + q0; // nested masks: T0>=T1>=T2>=T3

  float o[6];
#pragma unroll
  for (int r = 0; r < 6; ++r) {
    float W = (key[r] >= thr[0]) ? Q0
            : (key[r] >= thr[1]) ? Q1
            : (key[r] >= thr[2]) ? Q2
            : (key[r] >= thr[3]) ? Q3 : 0.f;
    o[r] = e[r] * W;
  }

  float4 v4; v4.x = o[0]; v4.y = o[1]; v4.z = o[2]; v4.w = o[3];
  *(float4*)(outp + 4 * lane) = v4;
  float2 v2; v2.x = o[4]; v2.y = o[5];
  *(float2*)(outp + 128 + 2 * lane) = v2;
}

__global__ __launch_bounds__(BLOCK_T) void sparsify_topk_softmax_kernel(
    const float* __restrict__ attn,
    const float* __restrict__ w1p, const float* __restrict__ w2p,
    const float* __restrict__ w3p, const float* __restrict__ w4p,
    float* __restrict__ out, int nrows) {
  __shared__ __attribute__((aligned(16))) float smem[WPB][2][CROW];
  const int lane = threadIdx.x & 31;
  const int wv   = threadIdx.x >> 5;
  const float w1 = w1p[0], w2 = w2p[0], w3 = w3p[0], w4 = w4p[0];

  long long r = (long long)blockIdx.x * WPB + wv;
  const long long stride = (long long)gridDim.x * WPB;

  // Persistent wave loop with per-wave double-buffered async global->LDS copy.
  if (r < nrows) {
    const float* g = attn + r * CROW;
    async_copy_b128(g + 4 * lane,       &smem[wv][0][4 * lane]);
    async_copy_b64 (g + 128 + 2 * lane, &smem[wv][0][128 + 2 * lane]);
  }
  int b = 0;
  for (; r < nrows; r += stride) {
    const long long rn = r + stride;
    if (rn < nrows) {
      const float* g = attn + rn * CROW;
      async_copy_b128(g + 4 * lane,       &smem[wv][b ^ 1][4 * lane]);
      async_copy_b64 (g + 128 + 2 * lane, &smem[wv][b ^ 1][128 + 2 * lane]);
      wait_async_le2(); // current row's 2 copies complete (async loads retire in order)
    } else {
      wait_async_le0();
    }
    const float* buf = smem[wv][b];
    float4 t4 = *(const float4*)(buf + 4 * lane);
    float2 t2 = *(const float2*)(buf + 128 + 2 * lane);
    float a[6] = {t4.x, t4.y, t4.z, t4.w, t2.x, t2.y};
    process_row(a, out + r * CROW, lane, w1, w2, w3, w4);
    b ^= 1;
  }
}

extern "C" void kernel_launch(void* const* d_in, const int* in_sizes, int n_in,
                              void* d_out, int out_size, void* d_ws, size_t ws_size,
                              hipStream_t stream) {
  (void)in_sizes; (void)n_in; (void)d_ws; (void)ws_size;
  const float* attn = (const float*)d_in[0];
  const float* w1   = (const float*)d_in[1];
  const float* w2   = (const float*)d_in[2];
  const float* w3   = (const float*)d_in[3];
  const float* w4   = (const float*)d_in[4];
  float* out = (float*)d_out;

  const int nrows = out_size / CROW; // 98,304 for the reference shapes
  int blocks = 2048;                 // persistent grid; >=2 rows/wave for prefetch overlap
  const int maxb = (nrows + WPB - 1) / WPB;
  if (blocks > maxb) blocks = maxb;
  if (blocks < 1) blocks = 1;

  hipLaunchKernelGGL(sparsify_topk_softmax_kernel, dim3(blocks), dim3(BLOCK_T), 0, stream,
                     attn, w1, w2, w3, w4, out, nrows);
}